// NodeAttrsFromEdgeAttrs_42193758716366
// MI455X (gfx1250) — compile-verified
//
#include <hip/hip_runtime.h>

typedef __attribute__((ext_vector_type(2))) float v2f;
typedef __attribute__((ext_vector_type(8))) float v8f;

#define D_FEAT 32

// ---------------------------------------------------------------------------
// Kernel 1: zero the output sums and the per-node counts (we accumulate with
// atomics, and the harness does not re-poison between replays).
// ---------------------------------------------------------------------------
__global__ void NodeAttrs_zero_kernel(float* __restrict__ out, int out_n,
                                      float* __restrict__ counts, int n_nodes) {
  int i = blockIdx.x * blockDim.x + threadIdx.x;
  int total = out_n + n_nodes;
  if (i < total) {
    if (i < out_n) out[i] = 0.0f;
    else           counts[i - out_n] = 0.0f;
  }
}

// ---------------------------------------------------------------------------
// WMMA segmented-sum helpers.
//
// Window model: a wave keeps a 16-node window [seg_base, seg_base+15] of
// accumulated sums in two 16x16 f32 WMMA accumulators (features 0-15 and
// 16-31). A = one-hot selector (16 window rows x 4 edges), B = 4 edges x 16
// features of raw f32 data, D += A*B does the segmented reduction on the
// matrix pipe at full f32 precision.
// ---------------------------------------------------------------------------
__device__ __forceinline__ void wmma_chunk(const float* __restrict__ attrs,
                                           long long ebase, int c4, int lane,
                                           int s0, int s1, int s2, int s3,
                                           int seg_base, long long n_edges,
                                           v8f& acc0, v8f& acc1) {
  const int half = lane >> 4;      // 0: K={0,1}, 1: K={2,3}
  const int m    = lane & 15;      // A row (M) for this lane == B column (N)
  // Edges covered by this lane's two K slots.
  const long long ea = ebase + c4 + (half ? 2 : 0);
  const long long eb = ea + 1;
  const int da = half ? s2 : s0;
  const int db = half ? s3 : s1;

  v2f A, B0, B1;
  A.x = (da - seg_base == m) ? 1.0f : 0.0f;
  A.y = (db - seg_base == m) ? 1.0f : 0.0f;

  const bool va = ea < n_edges;
  const bool vb = eb < n_edges;
  const float* ra = attrs + ea * D_FEAT;
  const float* rb = attrs + eb * D_FEAT;
  B0.x = va ? ra[m]      : 0.0f;
  B0.y = vb ? rb[m]      : 0.0f;
  B1.x = va ? ra[m + 16] : 0.0f;
  B1.y = vb ? rb[m + 16] : 0.0f;

  // D(16 nodes x 16 feats) += A(16x4 selector) * B(4 edges x 16 feats)
  acc0 = __builtin_amdgcn_wmma_f32_16x16x4_f32(false, A, false, B0,
                                               (short)0, acc0, false, false);
  acc1 = __builtin_amdgcn_wmma_f32_16x16x4_f32(false, A, false, B1,
                                               (short)0, acc1, false, false);
}

// Flush window rows [seg_base, win_max] with predicated float atomics, then
// clear the accumulators. C/D layout: VGPR j holds row M=j (lanes 0-15) and
// M=j+8 (lanes 16-31); column N = lane&15.
__device__ __forceinline__ void flush_window(float* __restrict__ out, int lane,
                                             int seg_base, int win_max,
                                             int n_nodes, v8f& acc0, v8f& acc1) {
  const int mhi  = (lane >> 4) * 8;
  const int feat = lane & 15;
#pragma unroll
  for (int j = 0; j < 8; ++j) {
    const int node = seg_base + mhi + j;
    if (node <= win_max && node < n_nodes) {
      atomicAdd(out + (long long)node * D_FEAT + feat,      acc0[j]);
      atomicAdd(out + (long long)node * D_FEAT + 16 + feat, acc1[j]);
    }
  }
  const v8f z = {0.f, 0.f, 0.f, 0.f, 0.f, 0.f, 0.f, 0.f};
  acc0 = z;
  acc1 = z;
}

// ---------------------------------------------------------------------------
// Kernel 2: grid-strided waves, one 32-edge x 32-feature tile per iteration.
// ---------------------------------------------------------------------------
__global__ void NodeAttrsFromEdgeAttrs_42193758716366_kernel(
    const float* __restrict__ attrs, const int* __restrict__ dst,
    float* __restrict__ sums, float* __restrict__ counts,
    long long n_edges, int n_nodes) {
  const int lane = threadIdx.x & 31;
  const long long wave   = (long long)blockIdx.x * (blockDim.x >> 5) + (threadIdx.x >> 5);
  const long long nwaves = (long long)gridDim.x * (blockDim.x >> 5);
  const long long ntiles = (n_edges + 31) >> 5;

  for (long long tile = wave; tile < ntiles; tile += nwaves) {
    const long long ebase = tile << 5;

    // Prefetch this wave's next tile (4KB: 128B per lane) into the caches.
    const long long pnext = (tile + nwaves) << 5;
    if (pnext < n_edges) {
      __builtin_prefetch(attrs + pnext * D_FEAT + (long long)lane * D_FEAT, 0, 0);
    }

    const long long el  = ebase + lane;
    const bool      inr = el < n_edges;
    const long long elc = inr ? el : (n_edges - 1);
    const int       dl  = dst[elc];

    // --- per-node counts: one atomic per run of equal dst within the tile ---
    {
      const int  prev = __shfl_up(dl, 1, 32);
      const bool head = inr && ((lane == 0) || (prev != dl));
      const unsigned hm = __builtin_amdgcn_ballot_w32(head);
      const unsigned vm = __builtin_amdgcn_ballot_w32(inr);
      if (head) {
        const unsigned rest = hm & ~((2u << lane) - 1u);     // heads above me
        const int next = rest ? (__ffs(rest) - 1) : 32;
        const unsigned below = (next < 32) ? ((1u << next) - 1u) : 0xFFFFFFFFu;
        const unsigned runmask = below & ~((1u << lane) - 1u) & vm;
        atomicAdd(counts + dl, (float)__popc(runmask));
      }
    }

    // --- windowed WMMA segmented sum over the 32-edge tile ---
    v8f acc0 = {0.f, 0.f, 0.f, 0.f, 0.f, 0.f, 0.f, 0.f};
    v8f acc1 = {0.f, 0.f, 0.f, 0.f, 0.f, 0.f, 0.f, 0.f};
    int seg_base = __builtin_amdgcn_readlane(dl, 0);  // dst of first tile edge
    int win_max  = seg_base;

#pragma unroll
    for (int c = 0; c < 8; ++c) {
      const int c4 = c * 4;
      const int s0 = __builtin_amdgcn_readlane(dl, c4 + 0);
      const int s1 = __builtin_amdgcn_readlane(dl, c4 + 1);
      const int s2 = __builtin_amdgcn_readlane(dl, c4 + 2);
      const int s3 = __builtin_amdgcn_readlane(dl, c4 + 3);

      // Chunk does not fit the current 16-node window: accumulate the part
      // that does, flush, and slide the window forward. (Sorted dst => edges
      // already processed fall below seg_base and select to zero.)
      while (s3 - seg_base > 15) {
        if (s0 - seg_base <= 15) {
          win_max = max(win_max, s0);
          if (s1 - seg_base <= 15) win_max = max(win_max, s1);
          if (s2 - seg_base <= 15) win_max = max(win_max, s2);
          wmma_chunk(attrs, ebase, c4, lane, s0, s1, s2, s3, seg_base, n_edges,
                     acc0, acc1);
        }
        flush_window(sums, lane, seg_base, win_max, n_nodes, acc0, acc1);
        int nb = s3;                       // smallest dst beyond the window
        if      (s0 - seg_base > 15) nb = s0;
        else if (s1 - seg_base > 15) nb = s1;
        else if (s2 - seg_base > 15) nb = s2;
        seg_base = nb;
        win_max  = seg_base - 1;           // window empty
      }

      wmma_chunk(attrs, ebase, c4, lane, s0, s1, s2, s3, seg_base, n_edges,
                 acc0, acc1);
      win_max = max(win_max, s3);
    }
    flush_window(sums, lane, seg_base, win_max, n_nodes, acc0, acc1);
  }
}

// ---------------------------------------------------------------------------
// Kernel 3: divide sums by clamped counts.
// ---------------------------------------------------------------------------
__global__ void NodeAttrs_finalize_kernel(float* __restrict__ out,
                                          const float* __restrict__ counts,
                                          long long total) {
  long long i = (long long)blockIdx.x * blockDim.x + threadIdx.x;
  if (i < total) {
    const float c = counts[i >> 5];  // D_FEAT == 32
    out[i] = out[i] / fmaxf(c, 1.0f);
  }
}

// ---------------------------------------------------------------------------
extern "C" void kernel_launch(void* const* d_in, const int* in_sizes, int n_in,
                              void* d_out, int out_size, void* d_ws, size_t ws_size,
                              hipStream_t stream) {
  const float* attrs = (const float*)d_in[0];
  const int*   dst   = (const int*)d_in[1];
  const long long n_edges = (long long)in_sizes[1];
  const int n_nodes = out_size / D_FEAT;

  float* out    = (float*)d_out;
  float* counts = (float*)d_ws;          // n_nodes floats of scratch

  // 1) zero sums + counts
  {
    const int threads = 256;
    const int total = out_size + n_nodes;
    const int blocks = (total + threads - 1) / threads;
    NodeAttrs_zero_kernel<<<blocks, threads, 0, stream>>>(out, out_size,
                                                          counts, n_nodes);
  }

  // 2) WMMA windowed scatter-add (grid-strided persistent waves)
  {
    const int threads = 256;               // 8 waves / block
    const int blocks  = 1024;              // 8192 waves total
    NodeAttrsFromEdgeAttrs_42193758716366_kernel<<<blocks, threads, 0, stream>>>(
        attrs, dst, out, counts, n_edges, n_nodes);
  }

  // 3) mean = sum / max(count, 1)
  {
    const int threads = 256;
    const long long total = (long long)out_size;
    const int blocks = (int)((total + threads - 1) / threads);
    NodeAttrs_finalize_kernel<<<blocks, threads, 0, stream>>>(out, counts, total);
  }
}